// EvoFill_77206332113541
// MI455X (gfx1250) — compile-verified
//
#include <hip/hip_runtime.h>
#include <hip/hip_bf16.h>

typedef float v2f __attribute__((ext_vector_type(2)));
typedef float v8f __attribute__((ext_vector_type(8)));

#define N_CATS 4
#define CHUNKL 512
#define OVERLAP 64
#define STEPL 448
#define D_MODEL 256
#define N_LAYERS 4
#define D_STATE 128
#define HEADDIM 64
#define D_INNER 512
#define NH 8
#define CONV_DIM 768
#define D_IN_PROJ 1288
#define NB 2
#define LSEQ 4096
#define NCHUNKS 9
#define NCG 18               // NB * NCHUNKS
#define MROWS (NCG * CHUNKL) // 9216

// CDNA5 async global->LDS copy (ASYNCcnt-tracked); vdst = LDS byte address.
__device__ __forceinline__ void async_ld16(void* lds, const void* g) {
  asm volatile("global_load_async_to_lds_b128 %0, %1, off"
               :: "v"((unsigned)(size_t)lds), "v"(g)
               : "memory");
}

// ---------------------------------------------------------------------------
// Embedding + coord projection + LayerNorm.  One block per (b,l), thread = d.
// ---------------------------------------------------------------------------
__global__ void __launch_bounds__(256)
embed_ln_kernel(const int* __restrict__ x, const float* __restrict__ coord,
                const float* __restrict__ table, const float* __restrict__ cw,
                const float* __restrict__ cb, const float* __restrict__ lw,
                const float* __restrict__ lb, float* __restrict__ out) {
  __shared__ float red[256];
  const int pos = blockIdx.x;            // b*LSEQ + l
  const int l = pos & (LSEQ - 1);
  const int d = threadIdx.x;
  const int xv = x[pos];
  const int idx = (xv == -1) ? N_CATS : xv;
  float e = table[idx * D_MODEL + d] + cb[d];
  #pragma unroll
  for (int j = 0; j < 4; ++j) e += coord[l * 4 + j] * cw[d * 4 + j];
  red[d] = e; __syncthreads();
  for (int s = 128; s > 0; s >>= 1) { if (d < s) red[d] += red[d + s]; __syncthreads(); }
  const float mean = red[0] / 256.0f;
  __syncthreads();
  const float c = e - mean;
  red[d] = c * c; __syncthreads();
  for (int s = 128; s > 0; s >>= 1) { if (d < s) red[d] += red[d + s]; __syncthreads(); }
  const float var = red[0] / 256.0f;
  out[(long)pos * D_MODEL + d] = c * rsqrtf(var + 1e-5f) * lw[d] + lb[d];
}

// ---------------------------------------------------------------------------
// Gather overlapping chunks: chunks[(b*9+c), t, d] = h[b, c*448+t, d]
// ---------------------------------------------------------------------------
__global__ void gather_chunks_kernel(const float* __restrict__ h, float* __restrict__ chunks) {
  const long i = (long)blockIdx.x * 256 + threadIdx.x;   // NCG*512*256 total
  const int d = (int)(i & 255);
  const int t = (int)((i >> 8) & 511);
  const int cg = (int)(i >> 17);
  const int b = cg / NCHUNKS, c = cg % NCHUNKS;
  chunks[i] = h[((long)b * LSEQ + c * STEPL + t) * D_MODEL + d];
}

// ---------------------------------------------------------------------------
// fp32 WMMA GEMM:  C[M,N] = A[M,K] * W[N,K]^T   (row-major A, W)
// 256 threads (8 waves), tile 64(M) x 32(N), K staged 32 at a time into a
// double-buffered LDS tile.  AMODE 0 uses global_load_async_to_lds_b128 so
// the DMA of K-tile k+1 overlaps the 8 V_WMMA_F32_16X16X4_F32 of tile k.
// AMODE 1: im2col for k=3 'same' conv (needs zero-fill) -> sync float4 path.
// GELU: fuse bias + exact GELU on store.
// ---------------------------------------------------------------------------
template <int AMODE, bool GELU>
__global__ void __launch_bounds__(256)
gemm_wmma_kernel(const float* __restrict__ A, const float* __restrict__ W,
                 const float* __restrict__ bias, float* __restrict__ C,
                 int M, int N, int K, int Lctx) {
  constexpr bool ASYNC = (AMODE == 0);
  __shared__ float As[2][64][36];   // 36-stride: 16B-aligned rows, bank-spread
  __shared__ float Bs[2][32][36];
  const int tid = threadIdx.x;
  const int n0 = blockIdx.x * 32;
  const int m0 = blockIdx.y * 64;
  const int wave = tid >> 5;
  const int lane = tid & 31;
  const int tm = (wave >> 1) * 16;       // 0,16,32,48
  const int tn = (wave & 1) * 16;        // 0,16
  // staging coords (float4 granularity; coalesced: consecutive tid -> +16B)
  const int ar = tid >> 3;               // 0..31 (+32 on second pass)
  const int ac = (tid & 7) * 4;          // 0..28
  const int bn = n0 + ar;
  const int bnc = (bn < N) ? bn : (N - 1);   // clamp: garbage cols discarded

  v8f acc = {};

  auto stage = [&](int sbuf, int k0) {
    if (ASYNC) {
      #pragma unroll
      for (int i = 0; i < 2; ++i) {
        const int r = ar + i * 32;
        async_ld16(&As[sbuf][r][ac], A + (long)(m0 + r) * K + k0 + ac);
      }
      async_ld16(&Bs[sbuf][ar][ac], W + (long)bnc * K + k0 + ac);
    } else {
      float4 va[2];
      #pragma unroll
      for (int i = 0; i < 2; ++i) {
        const int r = m0 + ar + i * 32;
        if (AMODE == 0) {
          va[i] = *(const float4*)(A + (long)r * K + k0 + ac);
        } else {
          const int kk = k0 + ac;
          const int tap = kk >> 8;       // 0..2
          const int ich = kk & 255;
          const int b = r / Lctx;
          const int l = r - b * Lctx;
          const int ls = l + tap - 1;
          va[i] = make_float4(0.f, 0.f, 0.f, 0.f);
          if (ls >= 0 && ls < Lctx)
            va[i] = *(const float4*)(A + ((long)b * Lctx + ls) * 256 + ich);
        }
      }
      const float4 vb = *(const float4*)(W + (long)bnc * K + k0 + ac);
      #pragma unroll
      for (int i = 0; i < 2; ++i)
        *(float4*)&As[sbuf][ar + i * 32][ac] = va[i];
      *(float4*)&Bs[sbuf][ar][ac] = vb;
    }
  };

  const int nblocks = K >> 5;
  stage(0, 0);
  for (int kb = 0; kb < nblocks; ++kb) {
    const int buf = kb & 1;
    if (kb + 1 < nblocks) {
      stage(buf ^ 1, (kb + 1) << 5);     // overlaps compute below (ASYNC)
      if (ASYNC) asm volatile("s_wait_asynccnt 0x3" ::: "memory");
    } else {
      if (ASYNC) asm volatile("s_wait_asynccnt 0x0" ::: "memory");
    }
    __syncthreads();
    // fp32 WMMA A layout: lane = row M (lane&15), K pair = (lane>>4)*2+{0,1};
    // B symmetric with lane = col N.  Preload all fragments, then WMMA chain.
    const int row = lane & 15;
    const int kq = (lane >> 4) * 2;
    v2f av[8], bv[8];
    #pragma unroll
    for (int q = 0; q < 8; ++q) {
      av[q].x = As[buf][tm + row][q * 4 + kq];
      av[q].y = As[buf][tm + row][q * 4 + kq + 1];
      bv[q].x = Bs[buf][tn + row][q * 4 + kq];
      bv[q].y = Bs[buf][tn + row][q * 4 + kq + 1];
    }
    #pragma unroll
    for (int q = 0; q < 8; ++q)
      acc = __builtin_amdgcn_wmma_f32_16x16x4_f32(false, av[q], false, bv[q],
                                                  (short)0, acc, false, false);
    __syncthreads();
  }
  // C/D layout: VGPR r -> M = r (lanes 0-15) / r+8 (lanes 16-31), N = lane&15
  const int col = lane & 15;
  const int mhalf = (lane >> 4) * 8;
  #pragma unroll
  for (int r = 0; r < 8; ++r) {
    const int m = m0 + tm + mhalf + r;
    const int n = n0 + tn + col;
    if (n < N) {
      float v = acc[r];
      if (GELU) {
        v += bias[n];
        v = 0.5f * v * (1.0f + erff(v * 0.70710678118654752440f));
      }
      C[(long)m * N + n] = v;
    }
  }
}

// ---------------------------------------------------------------------------
// Depthwise conv (D_CONV=4, left-pad 3) + SiLU over direction-ordered xBC.
// out[c, t, ch] (direction time order), reads zxbcdt channels [512,1280).
// ---------------------------------------------------------------------------
__global__ void __launch_bounds__(256)
conv_silu_kernel(const float* __restrict__ zx, const float* __restrict__ cw,
                 const float* __restrict__ cb, float* __restrict__ out, int rev) {
  const long i = (long)blockIdx.x * 256 + threadIdx.x;  // NCG*512*768
  const int ch = (int)(i % CONV_DIM);
  const int t = (int)((i / CONV_DIM) & 511);
  const int c = (int)(i / ((long)CONV_DIM * CHUNKL));
  float acc = cb[ch];
  #pragma unroll
  for (int k = 0; k < 4; ++k) {
    const int tt = t - 3 + k;
    if (tt >= 0) {
      const int torig = rev ? (511 - tt) : tt;
      acc += cw[ch * 4 + k] * zx[((long)c * CHUNKL + torig) * D_IN_PROJ + D_INNER + ch];
    }
  }
  out[i] = acc / (1.0f + expf(-acc));   // SiLU
}

// ---------------------------------------------------------------------------
// Selective scan: one block per (cg, head). 256 threads: p = tid&63 owns
// n-range [32*(tid>>6), +32).  512 sequential steps, ONE barrier per step:
// B/C staging tile and reduction scratch are both double-buffered, and the
// next step's B/C + x are prefetched into registers above the FMA chain.
// Writes gated output (ys + D*xs) * silu(z) at ORIGINAL time index.
// ---------------------------------------------------------------------------
__global__ void __launch_bounds__(256)
scan_kernel(const float* __restrict__ xbc, const float* __restrict__ zx,
            const float* __restrict__ dt_bias, const float* __restrict__ A_log,
            const float* __restrict__ Dp, float* __restrict__ yg, int rev) {
  __shared__ float sBC[2][256];     // [0,128) = B_t, [128,256) = C_t
  __shared__ float sred[2][256];
  const int tid = threadIdx.x;
  const int cg = blockIdx.x >> 3;
  const int h = blockIdx.x & 7;
  const int p = tid & 63;
  const int nbase = (tid >> 6) * 32;
  float hs[32];
  #pragma unroll
  for (int j = 0; j < 32; ++j) hs[j] = 0.0f;
  const float negA = -expf(A_log[h]);
  const float Dh = Dp[h];
  const float dtb = dt_bias[h];
  const long cbase = (long)cg * CHUNKL;
  sBC[0][tid] = xbc[cbase * CONV_DIM + D_INNER + tid];
  float xv = xbc[cbase * CONV_DIM + h * HEADDIM + p];
  __syncthreads();
  for (int t = 0; t < CHUNKL; ++t) {
    const int buf = t & 1;
    float bc_next = 0.0f, xv_next = 0.0f;          // prefetch t+1
    if (t + 1 < CHUNKL) {
      const long bn = (cbase + t + 1) * CONV_DIM;
      bc_next = xbc[bn + D_INNER + tid];
      xv_next = xbc[bn + h * HEADDIM + p];
    }
    const int torig = rev ? (511 - t) : t;
    const long basez = (cbase + torig) * D_IN_PROJ;
    const float dtr = zx[basez + (D_INNER + CONV_DIM) + h] + dtb;
    const float dt = (dtr > 20.0f) ? dtr : log1pf(expf(dtr));   // softplus
    const float dA = expf(negA * dt);
    const float coef = dt * xv;
    float acc = 0.0f;
    #pragma unroll
    for (int j = 0; j < 32; ++j) {
      const float hv = fmaf(dA, hs[j], coef * sBC[buf][nbase + j]);
      hs[j] = hv;
      acc = fmaf(hv, sBC[buf][128 + nbase + j], acc);
    }
    sred[buf][tid] = acc;
    if (t + 1 < CHUNKL) sBC[buf ^ 1][tid] = bc_next;
    __syncthreads();
    if (tid < 64) {
      float y = sred[buf][tid] + sred[buf][tid + 64] +
                sred[buf][tid + 128] + sred[buf][tid + 192];
      y = fmaf(Dh, xv, y);
      const float z = zx[basez + h * HEADDIM + p];
      yg[(cbase + torig) * D_INNER + h * HEADDIM + p] =
          y * (z / (1.0f + expf(-z)));
    }
    xv = xv_next;
  }
}

// ---------------------------------------------------------------------------
// Per-direction RMS-norm (over 512 dims) then combine fwd+rev, in place on ygf
// ---------------------------------------------------------------------------
__global__ void __launch_bounds__(256)
norm_combine_kernel(float* __restrict__ ygf, const float* __restrict__ ygr,
                    const float* __restrict__ nwf, const float* __restrict__ nwr) {
  __shared__ float r0[256];
  __shared__ float r1[256];
  const long row = blockIdx.x;
  const int tid = threadIdx.x;
  float* f = ygf + row * D_INNER;
  const float* r = ygr + row * D_INNER;
  const float v0 = f[tid], v1 = f[tid + 256];
  const float u0 = r[tid], u1 = r[tid + 256];
  r0[tid] = v0 * v0 + v1 * v1;
  r1[tid] = u0 * u0 + u1 * u1;
  __syncthreads();
  for (int s = 128; s > 0; s >>= 1) {
    if (tid < s) { r0[tid] += r0[tid + s]; r1[tid] += r1[tid + s]; }
    __syncthreads();
  }
  const float rsf = rsqrtf(r0[0] / (float)D_INNER + 1e-5f);
  const float rsr = rsqrtf(r1[0] / (float)D_INNER + 1e-5f);
  f[tid]       = v0 * rsf * nwf[tid]       + u0 * rsr * nwr[tid];
  f[tid + 256] = v1 * rsf * nwf[tid + 256] + u1 * rsr * nwr[tid + 256];
}

// ---------------------------------------------------------------------------
// Un-chunk: average positions covered by multiple overlapping chunks.
// ---------------------------------------------------------------------------
__global__ void overlap_avg_kernel(const float* __restrict__ chunks, float* __restrict__ out) {
  const long i = (long)blockIdx.x * 256 + threadIdx.x;  // NB*LSEQ*256
  const int d = (int)(i & 255);
  const int l = (int)((i >> 8) & (LSEQ - 1));
  const int b = (int)(i >> 20);
  float acc = 0.0f; int cnt = 0;
  int clo = (l - (CHUNKL - 1) + (STEPL - 1)) / STEPL; if (clo < 0) clo = 0;
  int chi = l / STEPL; if (chi > NCHUNKS - 1) chi = NCHUNKS - 1;
  for (int c = clo; c <= chi; ++c) {
    acc += chunks[(((long)b * NCHUNKS + c) * CHUNKL + (l - c * STEPL)) * D_MODEL + d];
    ++cnt;
  }
  out[i] = acc / (float)cnt;
}

// Pre-transpose lp_w (O,I,3) -> Wc[o][tap*256 + i] for the im2col GEMM.
__global__ void prep_wconv_kernel(const float* __restrict__ lpw, float* __restrict__ wc) {
  const int i = blockIdx.x * 256 + threadIdx.x;   // 256*768 exact
  const int o = i / 768;
  const int kk = i - o * 768;
  const int tap = kk >> 8, ich = kk & 255;
  wc[i] = lpw[(o * 256 + ich) * 3 + tap];
}

// ---------------------------------------------------------------------------
// Final 4-way logits: one wave per (b,l); lane-parallel over d, shuffle reduce
// ---------------------------------------------------------------------------
__global__ void __launch_bounds__(256)
logits_kernel(const float* __restrict__ cg, const float* __restrict__ ocw,
              const float* __restrict__ ocb, float* __restrict__ out) {
  const int wave = threadIdx.x >> 5;
  const int lane = threadIdx.x & 31;
  const int pos = blockIdx.x * 8 + wave;
  const float* v = cg + (long)pos * D_MODEL;
  float a0 = 0.f, a1 = 0.f, a2 = 0.f, a3 = 0.f;
  #pragma unroll
  for (int i = 0; i < 8; ++i) {
    const int d = lane + i * 32;
    const float xv = v[d];
    a0 = fmaf(xv, ocw[0 * D_MODEL + d], a0);
    a1 = fmaf(xv, ocw[1 * D_MODEL + d], a1);
    a2 = fmaf(xv, ocw[2 * D_MODEL + d], a2);
    a3 = fmaf(xv, ocw[3 * D_MODEL + d], a3);
  }
  #pragma unroll
  for (int s = 16; s > 0; s >>= 1) {
    a0 += __shfl_xor(a0, s, 32);
    a1 += __shfl_xor(a1, s, 32);
    a2 += __shfl_xor(a2, s, 32);
    a3 += __shfl_xor(a3, s, 32);
  }
  if (lane == 0) {
    out[pos * 4 + 0] = a0 + ocb[0];
    out[pos * 4 + 1] = a1 + ocb[1];
    out[pos * 4 + 2] = a2 + ocb[2];
    out[pos * 4 + 3] = a3 + ocb[3];
  }
}

// ---------------------------------------------------------------------------
extern "C" void kernel_launch(void* const* d_in, const int* in_sizes, int n_in,
                              void* d_out, int out_size, void* d_ws, size_t ws_size,
                              hipStream_t stream) {
  (void)in_sizes; (void)n_in; (void)out_size; (void)ws_size;
  const int*   x     = (const int*)d_in[0];
  const float* coord = (const float*)d_in[1];
  const float* table = (const float*)d_in[2];
  const float* cw    = (const float*)d_in[3];
  const float* cbv   = (const float*)d_in[4];
  const float* lnw   = (const float*)d_in[5];
  const float* lnb   = (const float*)d_in[6];
  const float* lpw   = (const float*)d_in[63];
  const float* lpb   = (const float*)d_in[64];
  const float* ocw   = (const float*)d_in[65];
  const float* ocb   = (const float*)d_in[66];

  float* ws   = (float*)d_ws;
  float* hbuf = ws;                               // 2,097,152
  float* chA  = hbuf + 2097152;                   // 2,359,296
  float* chB  = chA + 2359296;                    // 2,359,296
  float* zx   = chB + 2359296;                    // 11,870,208
  float* xbc  = zx + 11870208;                    // 7,077,888
  float* ygf  = xbc + 7077888;                    // 4,718,592
  float* ygr  = ygf + 4718592;                    // 4,718,592
  float* obuf = ygr + 4718592;                    // 2,097,152
  float* wc   = obuf + 2097152;                   // 196,608
  float* cgel = hbuf;                             // reuse embed buffer

  prep_wconv_kernel<<<768, 256, 0, stream>>>(lpw, wc);
  embed_ln_kernel<<<NB * LSEQ, 256, 0, stream>>>(x, coord, table, cw, cbv, lnw, lnb, hbuf);
  gather_chunks_kernel<<<(NCG * CHUNKL * D_MODEL) / 256, 256, 0, stream>>>(hbuf, chA);

  float* cur = chA;
  float* nxt = chB;
  for (int layer = 0; layer < N_LAYERS; ++layer) {
    const int base = 7 + layer * 14;
    const float* ipw = (const float*)d_in[base + 0];
    const float* opw = (const float*)d_in[base + 1];
    // shared in_proj (direction-independent, pointwise in t)
    gemm_wmma_kernel<0, false><<<dim3((D_IN_PROJ + 31) / 32, MROWS / 64), 256, 0, stream>>>(
        cur, ipw, nullptr, zx, MROWS, D_IN_PROJ, D_MODEL, 0);
    for (int dir = 0; dir < 2; ++dir) {
      const float* cvw  = (const float*)d_in[base + 2 + dir * 6];
      const float* cvb  = (const float*)d_in[base + 3 + dir * 6];
      const float* dtb  = (const float*)d_in[base + 4 + dir * 6];
      const float* alog = (const float*)d_in[base + 5 + dir * 6];
      const float* Dp   = (const float*)d_in[base + 6 + dir * 6];
      conv_silu_kernel<<<(NCG * CHUNKL * CONV_DIM) / 256, 256, 0, stream>>>(
          zx, cvw, cvb, xbc, dir);
      scan_kernel<<<NCG * NH, 256, 0, stream>>>(xbc, zx, dtb, alog, Dp,
                                                dir ? ygr : ygf, dir);
    }
    const float* nwf = (const float*)d_in[base + 7];
    const float* nwr = (const float*)d_in[base + 13];
    norm_combine_kernel<<<MROWS, 256, 0, stream>>>(ygf, ygr, nwf, nwr);
    // shared out_proj on (normed_fwd + unreversed normed_rev)
    gemm_wmma_kernel<0, false><<<dim3(D_MODEL / 32, MROWS / 64), 256, 0, stream>>>(
        ygf, opw, nullptr, nxt, MROWS, D_MODEL, D_INNER, 0);
    float* t = cur; cur = nxt; nxt = t;
  }

  overlap_avg_kernel<<<NB * LSEQ, 256, 0, stream>>>(cur, obuf);
  gemm_wmma_kernel<1, true><<<dim3(D_MODEL / 32, (NB * LSEQ) / 64), 256, 0, stream>>>(
      obuf, wc, lpb, cgel, NB * LSEQ, D_MODEL, 3 * D_MODEL, LSEQ);
  logits_kernel<<<(NB * LSEQ) / 8, 256, 0, stream>>>(cgel, ocw, ocb, (float*)d_out);
}